// LoRALinear_40355512714072
// MI455X (gfx1250) — compile-verified
//
#include <hip/hip_runtime.h>
#include <hip/hip_bf16.h>

// ---------------------------------------------------------------------------
// LoRA linear for MI455X (gfx1250, wave32, WMMA + Tensor Data Mover).
// out = x @ W^T + bias + SCALING * ((x @ A^T) @ B^T)
// Memory-bound at fp32 (~600MB min traffic @ 23.3TB/s ~ 26us vs ~7us of
// fp32-WMMA compute) -> full fp32 precision, V_WMMA_F32_16X16X4_F32, and
// TDM async tile DMA (global->LDS, double buffered) to overlap traffic
// with compute.
// ---------------------------------------------------------------------------

typedef __attribute__((ext_vector_type(2))) float v2f;
typedef __attribute__((ext_vector_type(8))) float v8f;
typedef __attribute__((ext_vector_type(4))) unsigned int u32x4;
typedef __attribute__((ext_vector_type(8))) unsigned int u32x8;

#define M_TOTAL 16384   // 4 * 4096
#define N_TOTAL 4096
#define K_TOTAL 4096
#define RANK    16
#define SCALING 2.0f    // 32.0 / 16

#define BM  128
#define BN  128
#define BK  32
#define LDK 40          // padded LDS k-stride (floats); TDM pads 8 DWORDs/row
#define NCHUNK (K_TOTAL / BK)   // 128

// ---------------------------------------------------------------------------
// Kernel 1: XR[m, r] = sum_k x[m,k] * lora_A[r,k]      (XR: [16384, 16] in ws)
// ---------------------------------------------------------------------------
__global__ __launch_bounds__(256)
void lora_xr_kernel(const float* __restrict__ x,
                    const float* __restrict__ loraA,
                    float* __restrict__ xr) {
    const int m    = blockIdx.x;
    const int tid  = threadIdx.x;
    const int lane = tid & 31;
    const int wid  = tid >> 5;

    const float* xrow = x + (size_t)m * K_TOTAL;

    float acc[RANK];
#pragma unroll
    for (int r = 0; r < RANK; ++r) acc[r] = 0.0f;

    for (int k = tid; k < K_TOTAL; k += 256) {
        float xv = xrow[k];
#pragma unroll
        for (int r = 0; r < RANK; ++r)
            acc[r] += xv * loraA[r * K_TOTAL + k];
    }

#pragma unroll
    for (int off = 16; off >= 1; off >>= 1) {
#pragma unroll
        for (int r = 0; r < RANK; ++r)
            acc[r] += __shfl_down(acc[r], off, 32);
    }

    __shared__ float partial[8][RANK];
    if (lane == 0) {
#pragma unroll
        for (int r = 0; r < RANK; ++r) partial[wid][r] = acc[r];
    }
    __syncthreads();

    if (tid < RANK) {
        float s = 0.0f;
#pragma unroll
        for (int w = 0; w < 8; ++w) s += partial[w][tid];
        xr[(size_t)m * RANK + tid] = s;
    }
}

// ---------------------------------------------------------------------------
// TDM tile load: 2D tensor (rows x K_TOTAL fp32), tile 128 rows x 32 cols,
// with LDS padding of 8 DWORDs after every 32-DWORD row (-> LDK=40 layout).
// Issued once per wave (EXEC ignored by TDM); completion via TENSORcnt.
// ---------------------------------------------------------------------------
__device__ __forceinline__
void tdm_load_tile(const float* gsrc, unsigned ldsAddr, unsigned tensorRows) {
    const unsigned long long ga = (unsigned long long)(uintptr_t)gsrc;

    u32x4 g0;
    g0[0] = 1u;                                        // count=1 user D#
    g0[1] = ldsAddr;                                   // LDS byte address
    g0[2] = (unsigned)(ga & 0xFFFFFFFFu);              // global_addr[31:0]
    g0[3] = (unsigned)((ga >> 32) & 0x01FFFFFFu)       // global_addr[56:32]
          | (2u << 30);                                // type = 2 ("image")

    u32x8 g1;
    g1[0] = (2u << 16)        // data_size = 4 bytes
          | (1u << 20)        // pad_enable
          | (4u << 22)        // pad_interval code 4 -> every 32 DWORDs
          | (7u << 25);       // pad_amount  code 7 -> 8 DWORDs of pad
    g1[1] = (K_TOTAL & 0xFFFFu) << 16;                         // tensor_dim0 lo
    g1[2] = (K_TOTAL >> 16) | ((tensorRows & 0xFFFFu) << 16);  // dim0 hi | dim1 lo
    g1[3] = (tensorRows >> 16) | ((unsigned)BK << 16);         // dim1 hi | tile_dim0
    g1[4] = (unsigned)BM;                                      // tile_dim1 (rows)
    g1[5] = (unsigned)K_TOTAL;                                 // dim0_stride lo
    g1[6] = 0u;                                                // stride hi / dim1_stride
    g1[7] = 0u;

    // 2-group form: VADDR2/VADDR3 = NULL (2D tensor).
    asm volatile("tensor_load_to_lds %0, %1" :: "s"(g0), "s"(g1) : "memory");
}

// ---------------------------------------------------------------------------
// Kernel 2: tiled WMMA GEMM  out = X @ W^T  (+ bias + SCALING * XR @ B^T)
// 256 threads = 8 waves (2 M x 4 N); block tile 128x128; K-chunk 32;
// wave tile 64x32 -> 4x2 v8f accumulators; TDM double-buffered LDS tiles.
// ---------------------------------------------------------------------------
__global__ __launch_bounds__(256)
void lora_gemm_kernel(const float* __restrict__ X,
                      const float* __restrict__ W,
                      const float* __restrict__ bias,
                      const float* __restrict__ loraB,
                      const float* __restrict__ XR,
                      float* __restrict__ out) {
    __shared__ float Xs[2][BM * LDK];
    __shared__ float Ws[2][BN * LDK];
    __shared__ float XRs[BM * RANK];
    __shared__ float Bs[BN * RANK];
    __shared__ float biasS[BN];

    const int tid   = threadIdx.x;
    const int lane  = tid & 31;
    const int wid   = tid >> 5;
    const int waveM = wid >> 2;     // 0..1 -> 64-row slab
    const int waveN = wid & 3;      // 0..3 -> 32-col slab
    const int half  = lane >> 4;    // lanes 16..31: +2 in K (A/B), +8 in M (C/D)
    const int l16   = lane & 15;

    const int n0 = blockIdx.x * BN;
    const int m0 = blockIdx.y * BM;

    // Stage LoRA epilogue operands and bias into LDS (covered by 1st barrier).
    for (int i = tid; i < BM * RANK; i += 256) XRs[i] = XR[(size_t)m0 * RANK + i];
    for (int i = tid; i < BN * RANK; i += 256) Bs[i]  = loraB[(size_t)n0 * RANK + i];
    if (tid < BN) biasS[tid] = bias[n0 + tid];

    v8f acc[4][2];
#pragma unroll
    for (int tm = 0; tm < 4; ++tm)
#pragma unroll
        for (int tn = 0; tn < 2; ++tn)
#pragma unroll
            for (int v = 0; v < 8; ++v) acc[tm][tn][v] = 0.0f;

    const float* Xblk = X + (size_t)m0 * K_TOTAL;
    const float* Wblk = W + (size_t)n0 * K_TOTAL;
    const unsigned xsAddr[2] = { (unsigned)(uintptr_t)(void*)&Xs[0][0],
                                 (unsigned)(uintptr_t)(void*)&Xs[1][0] };
    const unsigned wsAddr[2] = { (unsigned)(uintptr_t)(void*)&Ws[0][0],
                                 (unsigned)(uintptr_t)(void*)&Ws[1][0] };

    // Prologue: wave 0 issues chunk 0 DMA into buffer 0.
    if (wid == 0) {
        tdm_load_tile(Xblk, xsAddr[0], (unsigned)M_TOTAL);
        tdm_load_tile(Wblk, wsAddr[0], (unsigned)N_TOTAL);
    }

    for (int i = 0; i < NCHUNK; ++i) {
        const int cur = i & 1;

        __syncthreads();   // all waves done reading buf[(i+1)&1] (chunk i-1)

        if (wid == 0) {
            if (i + 1 < NCHUNK) {
                const int koff = (i + 1) * BK;
                tdm_load_tile(Xblk + koff, xsAddr[cur ^ 1], (unsigned)M_TOTAL);
                tdm_load_tile(Wblk + koff, wsAddr[cur ^ 1], (unsigned)N_TOTAL);
                // In-order TDM: waiting <=2 means chunk i's 2 tiles are done.
                __builtin_amdgcn_s_wait_tensorcnt(2);
            } else {
                __builtin_amdgcn_s_wait_tensorcnt(0);
            }
        }

        __syncthreads();   // chunk i tiles visible to all waves

        const float* xs = &Xs[cur][0];
        const float* ws = &Ws[cur][0];
#pragma unroll
        for (int kk = 0; kk < BK; kk += 4) {
            // B fragment (4x16, K-major): lanes 0-15 K=kk+{0,1}; lanes 16-31 +2.
            v2f bfrag[2];
#pragma unroll
            for (int tn = 0; tn < 2; ++tn) {
                const int n = waveN * 32 + tn * 16 + l16;
                bfrag[tn] = *(const v2f*)(&ws[n * LDK + kk + 2 * half]);
            }
            // A fragment (16x4): lanes 0-15 M=m, K=kk+{0,1}; lanes 16-31 +2.
            v2f afrag[4];
#pragma unroll
            for (int tm = 0; tm < 4; ++tm) {
                const int m = waveM * 64 + tm * 16 + l16;
                afrag[tm] = *(const v2f*)(&xs[m * LDK + kk + 2 * half]);
            }
#pragma unroll
            for (int tm = 0; tm < 4; ++tm)
#pragma unroll
                for (int tn = 0; tn < 2; ++tn)
                    acc[tm][tn] = __builtin_amdgcn_wmma_f32_16x16x4_f32(
                        false, afrag[tm], false, bfrag[tn],
                        (short)0, acc[tm][tn], false, false);
        }
    }

    // Epilogue: bias + SCALING * (XR @ B^T), then store.
    // C/D layout: VGPR v, lanes 0-15 -> M=v, N=lane; lanes 16-31 -> M=v+8.
#pragma unroll
    for (int tm = 0; tm < 4; ++tm) {
#pragma unroll
        for (int tn = 0; tn < 2; ++tn) {
            const int nloc = waveN * 32 + tn * 16 + l16;
            const int gn   = n0 + nloc;
            const float bv = biasS[nloc];
#pragma unroll
            for (int v = 0; v < 8; ++v) {
                const int mloc = waveM * 64 + tm * 16 + v + half * 8;
                float lora = 0.0f;
#pragma unroll
                for (int r = 0; r < RANK; ++r)
                    lora += XRs[mloc * RANK + r] * Bs[nloc * RANK + r];
                out[(size_t)(m0 + mloc) * N_TOTAL + gn] =
                    acc[tm][tn][v] + bv + SCALING * lora;
            }
        }
    }
}

// ---------------------------------------------------------------------------
extern "C" void kernel_launch(void* const* d_in, const int* in_sizes, int n_in,
                              void* d_out, int out_size, void* d_ws, size_t ws_size,
                              hipStream_t stream) {
    const float* x      = (const float*)d_in[0];   // [4,4096,4096]
    const float* weight = (const float*)d_in[1];   // [4096,4096]
    const float* bias   = (const float*)d_in[2];   // [4096]
    const float* loraA  = (const float*)d_in[3];   // [16,4096]
    const float* loraB  = (const float*)d_in[4];   // [4096,16]
    float* out = (float*)d_out;                    // [4,4096,4096]
    float* xr  = (float*)d_ws;                     // [16384,16] = 1 MB scratch

    // Phase 1: rank-16 projection XR = x @ A^T
    lora_xr_kernel<<<dim3(M_TOTAL), dim3(256), 0, stream>>>(x, loraA, xr);

    // Phase 2: fused WMMA GEMM + bias + LoRA epilogue (TDM double-buffered)
    dim3 grid(N_TOTAL / BN, M_TOTAL / BM);   // (32, 128)
    lora_gemm_kernel<<<grid, dim3(256), 0, stream>>>(x, weight, bias, loraB, xr, out);
}